// OneUnkMaterialStack_NK_79061757984793
// MI455X (gfx1250) — compile-verified
//
#include <hip/hip_runtime.h>
#include <hip/hip_bf16.h>
#include <math.h>

#define TWO_PI_F   6.28318530717958647692f
#define IM_CLAMP_F 35.0f

typedef __attribute__((ext_vector_type(2))) float v2f;
typedef __attribute__((ext_vector_type(8))) float v8f;

// ---------------- complex helpers (fp32, matches complex64 reference) ----------------
struct C2 { float x, y; };
__device__ __forceinline__ C2 mkc(float x, float y){ C2 c; c.x = x; c.y = y; return c; }
__device__ __forceinline__ C2 cadd(C2 a, C2 b){ return mkc(a.x + b.x, a.y + b.y); }
__device__ __forceinline__ C2 csub(C2 a, C2 b){ return mkc(a.x - b.x, a.y - b.y); }
__device__ __forceinline__ C2 cmul(C2 a, C2 b){
  return mkc(fmaf(a.x, b.x, -a.y * b.y), fmaf(a.x, b.y, a.y * b.x));
}
__device__ __forceinline__ C2 cscale(C2 a, float s){ return mkc(a.x * s, a.y * s); }
__device__ __forceinline__ C2 cinv(C2 a){
  float s = 1.0f / fmaf(a.x, a.x, a.y * a.y);
  return mkc(a.x * s, -a.y * s);
}
__device__ __forceinline__ C2 csqrtc(C2 z){           // principal branch
  float m  = sqrtf(fmaf(z.x, z.x, z.y * z.y));
  float re = sqrtf(fmaxf(0.5f * (m + z.x), 0.0f));
  float im = sqrtf(fmaxf(0.5f * (m - z.x), 0.0f));
  return mkc(re, copysignf(im, z.y));
}

// ---------------- CDNA5 async global->LDS staging (ASYNCcnt path) ----------------
__device__ __forceinline__ void async_g2l_b32(void* lds_ptr, const void* gptr){
  unsigned int        lds_off = (unsigned int)(size_t)lds_ptr;        // low 32 bits = DS offset
  unsigned long long  gaddr   = (unsigned long long)(size_t)gptr;
  asm volatile("global_load_async_to_lds_b32 %0, %1, off"
               :: "v"(lds_off), "v"(gaddr) : "memory");
}
__device__ __forceinline__ void wait_asynccnt0(){
  asm volatile("s_wait_asynccnt 0x0" ::: "memory");
}

// Kernel 0: pack Y[P][8] (cols = material-major n,k) and sin(theta) table.
__global__ void k_prep(const float* __restrict__ fixed_data,
                       const float* __restrict__ rin,
                       const float* __restrict__ kex,
                       const float* __restrict__ theta,
                       float* __restrict__ Y,
                       float* __restrict__ sin_out,
                       int P, int A, int NM)
{
  int g = blockIdx.x * blockDim.x + threadIdx.x;
  if (g < A) sin_out[g] = sinf(theta[g]);
  if (g >= P) return;
  for (int m = 0; m < NM && m < 4; ++m){
    float nv, kv;
    if (m < NM - 1){
      nv = fixed_data[(size_t)m * 3 * P + P     + g];
      kv = fixed_data[(size_t)m * 3 * P + 2 * P + g];
    } else {
      nv = rin[g]; kv = kex[g];
    }
    Y[(size_t)g * 8 + 2 * m + 0] = nv;
    Y[(size_t)g * 8 + 2 * m + 1] = kv;
  }
}

// Kernel 1: interpolation as a sparse-weight GEMM on the WMMA path.
//   nk[W x 8] = Wt[W x P] * Y[P x 8],  Wt row w = (1-t) at lo, t at hi.
// One wave32 per 16-row tile. The wave reduces the active K-window
// [min(lo), max(hi)] (all weights outside are zero) and broadcasts it to
// SGPRs, so the loop is scalar-bounded (EXEC provably all-1s at each WMMA)
// and only ~2-3 k-steps execute per tile. All loads are branchless
// (clamped addresses; unused D columns are never stored, zero-weight
// K-slots can't contribute).
__global__ __launch_bounds__(32)
void k_interp_wmma(const float* __restrict__ grid_xs,   // shared wavelength grid (P)
                   const float* __restrict__ Y,         // P x 8
                   const float* __restrict__ wl,        // W
                   float* __restrict__ nk_out,          // W x 8
                   int P, int W, int NM)
{
  const int  lane = threadIdx.x;       // 0..31, full wave active
  const int  tile = blockIdx.x;
  const int  rrow = lane & 15;
  const bool hi   = lane >= 16;

  int wr = tile * 16 + rrow; if (wr > W - 1) wr = W - 1;
  float x = wl[wr];

  // jnp.interp bracketing (clamped ends) on the shared grid
  int lo, hh; float t;
  if (x <= grid_xs[0])          { lo = 0;     hh = 0;     t = 0.f; }
  else if (x >= grid_xs[P - 1]) { lo = P - 1; hh = P - 1; t = 0.f; }
  else {
    int l = 0, h = P - 1;
    while (h - l > 1){ int mid = (l + h) >> 1; if (grid_xs[mid] <= x) l = mid; else h = mid; }
    lo = l; hh = h;
    t = (x - grid_xs[l]) / (grid_xs[h] - grid_xs[l]);
  }
  const float w_lo = 1.0f - t, w_hi = t;

  // wave-wide active K-window; broadcast to SGPRs -> uniform scalar loop bound
  int kmin = lo, kmax = hh;
#pragma unroll
  for (int off = 16; off > 0; off >>= 1){
    int tmn = __shfl_xor(kmin, off, 32);
    int tmx = __shfl_xor(kmax, off, 32);
    kmin = (tmn < kmin) ? tmn : kmin;
    kmax = (tmx > kmax) ? tmx : kmax;
  }
  const int skmin = __builtin_amdgcn_readfirstlane(kmin) & ~3;
  const int skmax = __builtin_amdgcn_readfirstlane(kmax);

  const int n = rrow;                              // output column this lane carries
  const float* Ycol = Y + ((n < 8) ? n : 7);       // clamped: garbage cols never stored

  v8f acc = {};
  for (int kb = skmin; kb <= skmax; kb += 4){
    // A 16x4 f32 layout: v0 = K(0|2), v1 = K(1|3) for lane halves (ISA 7.12.2)
    int k0 = kb + (hi ? 2 : 0);
    int k1 = kb + (hi ? 3 : 1);
    v2f a, b;
    a.x = (k0 == lo ? w_lo : 0.f) + (k0 == hh ? w_hi : 0.f);
    a.y = (k1 == lo ? w_lo : 0.f) + (k1 == hh ? w_hi : 0.f);
    // B 4x16 f32 mirror layout: v0 = row(0|2), v1 = row(1|3) for lane halves.
    // Clamped unconditional loads: rows >= P have zero A-weight in every lane.
    int c0 = (k0 < P - 1) ? k0 : P - 1;
    int c1 = (k1 < P - 1) ? k1 : P - 1;
    b.x = Ycol[(size_t)c0 * 8];
    b.y = Ycol[(size_t)c1 * 8];
    acc = __builtin_amdgcn_wmma_f32_16x16x4_f32(false, a, false, b, (short)0, acc,
                                                false, false);
  }

  // D 16x16 f32: vgpr r, lanes 0-15 -> M=r ; lanes 16-31 -> M=8+r
  if (n < 2 * NM){
    float* dst = nk_out + ((size_t)(tile * 16 + (hi ? 8 : 0)) * 8 + n);
    if (tile * 16 + 16 <= W){                 // full tile: unguarded stores
#pragma unroll
      for (int r = 0; r < 8; ++r) dst[(size_t)r * 8] = acc[r];
    } else {                                  // ragged last tile
#pragma unroll
      for (int r = 0; r < 8; ++r){
        int gw = tile * 16 + (hi ? 8 : 0) + r;
        if (gw < W) nk_out[(size_t)gw * 8 + n] = acc[r];
      }
    }
  }
}

// Kernel 2: full TMM chain. block = (one wavelength, up to 256 angles) = 8 wave32 waves.
__global__ __launch_bounds__(256)
void k_tmm(const float* __restrict__ nk_tab,
           const float* __restrict__ sin_tab,
           const float* __restrict__ wl,
           const int*   __restrict__ md,
           const float* __restrict__ th_above,
           const float* __restrict__ th_unk,
           const float* __restrict__ th_below,
           int na, int nu, int nb,
           const int* __restrict__ pol_p,
           float* __restrict__ out,
           int W, int A, int L, int NM)
{
  const int w   = blockIdx.x;
  const int tid = threadIdx.x;
  const int a   = blockIdx.y * blockDim.x + tid;

  __shared__ float s_nk[8];            // n,k per material (NM <= 4), uniform per block
  __shared__ float s_sin[256];         // sin(theta) for this angle chunk
  __shared__ int   s_md[32];           // layer -> material
  __shared__ float s_cr[4][256], s_ci[4][256];   // cos(theta_t) per material, per lane
  __shared__ float s_pr[4][256], s_pi[4][256];   // n*cos per material, per lane

  // ---- CDNA5 async staging: global -> LDS without touching the VGPR data path ----
  {
    const float* gnk = nk_tab + (size_t)w * 8;
    __builtin_prefetch(gnk, 0, 0);                       // global_prefetch_b8
    if (tid < 2 * NM && tid < 8) async_g2l_b32(&s_nk[tid], gnk + tid);
    if (a < A)                   async_g2l_b32(&s_sin[tid], sin_tab + a);
    if (tid < L && tid < 32)     async_g2l_b32(&s_md[tid], md + tid);
    wait_asynccnt0();
    __syncthreads();
  }
  if (a >= A) return;

  const float wlv  = wl[w];
  const float kfac = TWO_PI_F / wlv;
  const int   m0   = s_md[0];
  const C2    n0   = mkc(s_nk[2 * m0], s_nk[2 * m0 + 1]);
  const C2    sin0 = cscale(n0, s_sin[tid]);

  // per unique material: cos(theta_t) = sqrt(1 - (sin0/n)^2), ncos = n*cos
  for (int m = 0; m < NM && m < 4; ++m){
    C2 nm  = mkc(s_nk[2 * m], s_nk[2 * m + 1]);
    C2 st  = cmul(sin0, cinv(nm));
    C2 st2 = cmul(st, st);
    C2 ct  = csqrtc(mkc(1.0f - st2.x, -st2.y));
    C2 nc  = cmul(nm, ct);
    s_cr[m][tid] = ct.x; s_ci[m][tid] = ct.y;
    s_pr[m][tid] = nc.x; s_pi[m][tid] = nc.y;
  }

  // ---- chain: propagate first column of Mtot right-to-left; both pols share exp/sincos ----
  int mlast = s_md[L - 1];
  C2 ct_n = mkc(s_cr[mlast][tid], s_ci[mlast][tid]);   // layer j+1 ("next") values
  C2 nc_n = mkc(s_pr[mlast][tid], s_pi[mlast][tid]);
  C2 n_n  = mkc(s_nk[2 * mlast], s_nk[2 * mlast + 1]);

  C2 vs0 = mkc(1.f, 0.f), vs1 = mkc(0.f, 0.f);   // s-pol column
  C2 vp0 = mkc(1.f, 0.f), vp1 = mkc(0.f, 0.f);   // p-pol column
  C2 tps = mkc(1.f, 0.f), tpp = mkc(1.f, 0.f);   // running product of t's

  const int nd = na + nu + nb;
  for (int j = L - 2; j >= 1; --j){
    int mj = s_md[j];
    C2 ct_c = mkc(s_cr[mj][tid], s_ci[mj][tid]);
    C2 nc_c = mkc(s_pr[mj][tid], s_pi[mj][tid]);
    C2 n_c  = mkc(s_nk[2 * mj], s_nk[2 * mj + 1]);

    // interface j (between layer j and j+1)
    C2 idens = cinv(cadd(nc_c, nc_n));
    C2 rs    = cmul(csub(nc_c, nc_n), idens);
    C2 ts    = cmul(cscale(nc_c, 2.0f), idens);

    C2 ap    = cmul(n_n, ct_c);
    C2 bp    = cmul(n_c, ct_n);
    C2 idenp = cinv(cadd(ap, bp));
    C2 rp    = cmul(csub(ap, bp), idenp);
    C2 tp    = cmul(cscale(nc_c, 2.0f), idenp);   // 2*n_j*cos_j = 2*ncos_j

    // propagation phase of layer j (thickness index j-1), imag clamped
    int ti = j - 1; if (ti >= nd) ti = nd - 1; if (ti < 0) ti = 0;
    float d  = (ti < na) ? th_above[ti]
             : (ti < na + nu) ? th_unk[ti - na]
             : th_below[ti - na - nu];
    float dr = kfac * d * nc_c.x;
    float di = fminf(fmaxf(kfac * d * nc_c.y, -IM_CLAMP_F), IM_CLAMP_F);
    float sd, cd; sincosf(dr, &sd, &cd);
    float e  = expf(di);
    float ei = 1.0f / e;
    C2 em = mkc(e  * cd, -e  * sd);   // exp(-i*delta)
    C2 ep = mkc(ei * cd,  ei * sd);   // exp(+i*delta)

    // v' = P_j * (I_j * v)  (unscaled; 1/t folded into tprod)
    C2 a0 = cadd(vs0, cmul(rs, vs1));
    C2 a1 = cadd(cmul(rs, vs0), vs1);
    vs0 = cmul(em, a0); vs1 = cmul(ep, a1); tps = cmul(tps, ts);

    C2 b0 = cadd(vp0, cmul(rp, vp1));
    C2 b1 = cadd(cmul(rp, vp0), vp1);
    vp0 = cmul(em, b0); vp1 = cmul(ep, b1); tpp = cmul(tpp, tp);

    ct_n = ct_c; nc_n = nc_c; n_n = n_c;   // layer j becomes "next"
  }

  // interface 0 (ambient -> layer 1), no propagation matrix
  C2 ct0 = mkc(s_cr[m0][tid], s_ci[m0][tid]);
  C2 nc0 = mkc(s_pr[m0][tid], s_pi[m0][tid]);
  {
    C2 idens = cinv(cadd(nc0, nc_n));
    C2 rs    = cmul(csub(nc0, nc_n), idens);
    C2 ts    = cmul(cscale(nc0, 2.0f), idens);
    C2 a0 = cadd(vs0, cmul(rs, vs1));
    C2 a1 = cadd(cmul(rs, vs0), vs1);
    vs0 = a0; vs1 = a1; tps = cmul(tps, ts);

    C2 ap    = cmul(n_n, ct0);
    C2 bp    = cmul(n0, ct_n);
    C2 idenp = cinv(cadd(ap, bp));
    C2 rp    = cmul(csub(ap, bp), idenp);
    C2 tp    = cmul(cscale(nc0, 2.0f), idenp);
    C2 b0 = cadd(vp0, cmul(rp, vp1));
    C2 b1 = cadd(cmul(rp, vp0), vp1);
    vp0 = b0; vp1 = b1; tpp = cmul(tpp, tp);
  }

  // rr = M10/M00 (scale-free), tt = prod(t)/M00
  C2 ivs = cinv(vs0);
  C2 rrs = cmul(vs1, ivs);
  C2 tts = cmul(tps, ivs);
  C2 ivp = cinv(vp0);
  C2 rrp = cmul(vp1, ivp);
  C2 ttp = cmul(tpp, ivp);

  float clx = s_cr[mlast][tid], cly = s_ci[mlast][tid];
  float plx = s_pr[mlast][tid];
  float nlr = s_nk[2 * mlast], nli = s_nk[2 * mlast + 1];

  float Tfs = plx / nc0.x;                                                  // Re ncos ratio
  float Tfp = fmaf(nlr, clx, nli * cly) / fmaf(n0.x, ct0.x, n0.y * ct0.y);  // Re(n conj(cos)) ratio

  float Rs = fmaf(rrs.x, rrs.x, rrs.y * rrs.y);
  float Ts = fmaf(tts.x, tts.x, tts.y * tts.y) * Tfs;
  float Rp = fmaf(rrp.x, rrp.x, rrp.y * rrp.y);
  float Tp = fmaf(ttp.x, ttp.x, ttp.y * ttp.y) * Tfp;

  int pol = *pol_p;
  float R, T;
  if (pol == 0)      { R = Rs; T = Ts; }
  else if (pol == 1) { R = Rp; T = Tp; }
  else               { R = 0.5f * (Rs + Rp); T = 0.5f * (Ts + Tp); }

  size_t WA   = (size_t)W * A;
  size_t base = (size_t)w * A + a;
  out[base]          = R;
  out[WA + base]     = T;
  out[2 * WA + base] = 1.0f - (R + T);
}

extern "C" void kernel_launch(void* const* d_in, const int* in_sizes, int n_in,
                              void* d_out, int out_size, void* d_ws, size_t ws_size,
                              hipStream_t stream) {
  const float* fixed_data = (const float*)d_in[0];
  const float* rin        = (const float*)d_in[1];
  const float* kex        = (const float*)d_in[2];
  const float* dynwl      = (const float*)d_in[3];
  const float* th_above   = (const float*)d_in[4];
  const float* th_below   = (const float*)d_in[5];
  const float* th_unk     = (const float*)d_in[6];
  const float* wl         = (const float*)d_in[7];
  const float* theta      = (const float*)d_in[8];
  const int*   md         = (const int*)d_in[9];
  const int*   pol        = (const int*)d_in[10];

  int P  = in_sizes[3];
  int Mf = (P > 0) ? in_sizes[0] / (3 * P) : 3;
  int NM = Mf + 1; if (NM > 4) NM = 4;
  int na = in_sizes[4];
  int nb = in_sizes[5];
  int nu = in_sizes[6];
  int W  = in_sizes[7];
  int A  = in_sizes[8];
  int L  = in_sizes[9];

  // workspace: nk table [W x 8] | sin table [A] | Y [P x 8]
  float* nk_tab  = (float*)d_ws;
  float* sin_tab = nk_tab + (size_t)W * 8;
  float* Ytab    = sin_tab + (size_t)A;

  const float* grid_xs = (Mf >= 1) ? fixed_data : dynwl;   // shared wavelength grid

  int pa = (P > A) ? P : A;
  k_prep<<<(pa + 255) / 256, 256, 0, stream>>>(fixed_data, rin, kex, theta,
                                               Ytab, sin_tab, P, A, NM);

  int ntiles = (W + 15) / 16;
  k_interp_wmma<<<ntiles, 32, 0, stream>>>(grid_xs, Ytab, wl, nk_tab, P, W, NM);

  dim3 grid(W, (A + 255) / 256);
  // NOTE concat order in reference: [above, unknown, below]
  k_tmm<<<grid, 256, 0, stream>>>(nk_tab, sin_tab, wl, md,
                                  th_above, th_unk, th_below,
                                  na, nu, nb, pol, (float*)d_out, W, A, L, NM);
}